// DeltaModulationEncoder_52037823758588
// MI455X (gfx1250) — compile-verified
//
#include <hip/hip_runtime.h>
#include <hip/hip_bf16.h>
#include <stdint.h>

// Problem geometry (fixed by the reference): x is (16, 256, 8192) f32.
#define T_LEN   8192
#define NSEQ    4096              // 16 * 256 independent scan chains
#define ROWS    32                // sequences per block == lanes per wave (wave32)
#define TC      128               // timesteps staged per chunk
#define NCHUNK  (T_LEN / TC)      // 64 chunks
#define PADF    4                 // LDS row pad (floats): rows stay 16B aligned, breaks bank stride
#define ROWF    (TC + PADF)       // floats per LDS row (528 B row stride)

// ---- CDNA5 async global->LDS copy engine ------------------------------------
// One instruction moves 32 lanes x 16 B = 512 B = one full TC-float row of the tile.
// LDS byte offset = low 32 bits of the generic address of a __shared__ object
// (flat LDS aperture base has zero low bits per ISA 10.2 aperture layout).
__device__ __forceinline__ void async_tile_load(const float* __restrict__ x,
                                                uint32_t lds_base, int seqBase,
                                                int chunk, int lane) {
  const char* gbase = (const char*)x
                    + ((size_t)seqBase << 15)        // seqBase * 8192 floats * 4B
                    + (size_t)chunk * (TC * 4)
                    + (size_t)lane * 16;
  uint32_t lds = lds_base + (uint32_t)lane * 16u;
#pragma unroll
  for (int j = 0; j < ROWS; ++j) {                   // one async op per sequence row
    const void* g = gbase + (size_t)j * (T_LEN * 4);
    asm volatile("global_load_async_to_lds_b128 %0, %1, off"
                 :: "v"(lds), "v"(g) : "memory");
    lds += ROWF * 4;
  }
}

// Async loads retire in order, so <=32 outstanding implies the older chunk landed.
__device__ __forceinline__ void wait_async_prev_chunk() {
  asm volatile("s_wait_asynccnt 0x20" ::: "memory");
}
__device__ __forceinline__ void wait_async_all() {
  asm volatile("s_wait_asynccnt 0x0" ::: "memory");
}

// One delta-modulation step; bit-exact vs reference:
//   net*th == copysign(th, err) when |err|>th else 0;  (+-0.1f)*10.0f == +-1.0f exactly.
__device__ __forceinline__ float dm_step(float xv, float& recon) {
  const float th = 0.1f, inv_th = 10.0f;
  float err  = xv - recon;
  float mag  = (__builtin_fabsf(err) > th) ? th : 0.0f;  // cmp uses free |.| src modifier
  float step = __builtin_copysignf(mag, err);            // v_bfi
  recon += step;
  return step * inv_th;                                  // exactly {-1, 0(+-), 1}
}

__global__ __launch_bounds__(ROWS)
void DeltaModulationEncoder_kernel(const float* __restrict__ x,
                                   float* __restrict__ out) {
  __shared__ __align__(16) float tin[2][ROWS][ROWF];     // 33 KB of the 320 KB LDS

  const int lane    = threadIdx.x;                       // wave32: lane == thread
  const int seqBase = blockIdx.x * ROWS;
  const int seq     = seqBase + lane;

  const uint32_t lds0 = (uint32_t)(uintptr_t)&tin[0][0][0];
  const uint32_t lds1 = (uint32_t)(uintptr_t)&tin[1][0][0];

  // Prime the pipeline: chunk 0 -> buffer 0.
  async_tile_load(x, lds0, seqBase, 0, lane);

  float recon = 0.0f;
  float4* gout = (float4*)(out + (size_t)seq * T_LEN);

  for (int c = 0; c < NCHUNK; ++c) {
    const int buf = c & 1;
    if (c + 1 < NCHUNK) {
      async_tile_load(x, (buf ^ 1) ? lds1 : lds0, seqBase, c + 1, lane);
      wait_async_prev_chunk();   // chunk c resident; chunk c+1 still streaming in
    } else {
      wait_async_all();
    }

    const float4* inrow = (const float4*)&tin[buf][lane][0];  // 528B stride, 16B aligned
#pragma unroll 8
    for (int q = 0; q < TC / 4; ++q) {
      float4 v = inrow[q];                                    // ds_load_b128
      float4 s;
      s.x = dm_step(v.x, recon);
      s.y = dm_step(v.y, recon);
      s.z = dm_step(v.z, recon);
      s.w = dm_step(v.w, recon);
      gout[q] = s;                                            // global_store_b128
    }
    gout += TC / 4;
  }
}

extern "C" void kernel_launch(void* const* d_in, const int* in_sizes, int n_in,
                              void* d_out, int out_size, void* d_ws, size_t ws_size,
                              hipStream_t stream) {
  (void)in_sizes; (void)n_in; (void)d_ws; (void)ws_size; (void)out_size;
  const float* x = (const float*)d_in[0];
  float* out = (float*)d_out;
  // 4096 sequences, 32 per block -> 128 blocks of one wave32 each (spreads over WGPs).
  DeltaModulationEncoder_kernel<<<dim3(NSEQ / ROWS), dim3(ROWS), 0, stream>>>(x, out);
}